// OneStepDecoder_90142773608956
// MI455X (gfx1250) — compile-verified
//
#include <hip/hip_runtime.h>
#include <math.h>

// ---------------------------------------------------------------------------
// One-step attention + LSTM decoder for MI455X (gfx1250, wave32).
// f32 end-to-end (workload is HBM-bound at ~440 MB / 23.3 TB/s, so f32 WMMA
// keeps reference precision at zero bandwidth cost). GEMMs use
// V_WMMA_F32_16X16X4_F32 fed from LDS, which is double-buffered via the
// CDNA5 async global->LDS path (GLOBAL_LOAD_ASYNC_TO_LDS_B128 + ASYNCcnt).
// ---------------------------------------------------------------------------

typedef __attribute__((ext_vector_type(2))) float v2f;
typedef __attribute__((ext_vector_type(8))) float v8f;
typedef __attribute__((ext_vector_type(4))) int   v4i;

// AMDGPU address-space typed b128 pointers (clang prints AS1 as __device__)
typedef __attribute__((address_space(1))) v4i* as1_v4i_ptr;
typedef __attribute__((address_space(3))) v4i* as3_v4i_ptr;

#define B_  128
#define T_  256
#define D_  1024
#define V_  32000
#define E_  300
#define DE_ (D_ + E_)   // 1324 (divisible by 4)
#define G4_ (4 * D_)    // 4096

#define BN  512         // block N tile (8 waves x 64 cols)
#define BK  16          // K slab staged in LDS per stage
#define LDSB (BK * BN)  // floats per LDS buffer (32 KB)

// ------------------- CDNA5 async global->LDS primitives --------------------
__device__ __forceinline__ void async_copy16(const float* gsrc, float* lds_dst) {
#if defined(__HIP_DEVICE_COMPILE__)
#if __has_builtin(__builtin_amdgcn_global_load_async_to_lds_b128)
    __builtin_amdgcn_global_load_async_to_lds_b128(
        (as1_v4i_ptr)(size_t)gsrc,      // AS1 int4* (const dropped by cast)
        (as3_v4i_ptr)lds_dst,           // AS3 int4*
        0, 0);
#else
    unsigned loff = (unsigned)(unsigned long long)
        (__attribute__((address_space(3))) float*)lds_dst;
    asm volatile("global_load_async_to_lds_b128 %0, %1, off"
                 :: "v"(loff), "v"((unsigned long long)(size_t)gsrc)
                 : "memory");
#endif
#else
    (void)gsrc; (void)lds_dst;
#endif
}

__device__ __forceinline__ void wait_async0() {
#if defined(__HIP_DEVICE_COMPILE__)
#if __has_builtin(__builtin_amdgcn_s_wait_asynccnt)
    __builtin_amdgcn_s_wait_asynccnt(0);
#else
    asm volatile("s_wait_asynccnt 0" ::: "memory");
#endif
#endif
}

__device__ __forceinline__ float wave_sum(float v) {
    #pragma unroll
    for (int off = 16; off > 0; off >>= 1)
        v += __shfl_down(v, off, 32);
    return v;
}

// -------------------------- 1) embedding gather ----------------------------
__global__ void emb_gather_kernel(const int* __restrict__ idx,
                                  const float* __restrict__ table,
                                  float* __restrict__ x) {
    int i = blockIdx.x * blockDim.x + threadIdx.x;
    if (i >= B_ * E_) return;
    int b = i / E_;
    int e = i - b * E_;
    x[(size_t)b * DE_ + D_ + e] = table[(size_t)idx[b] * E_ + e];
}

// -------------------------- 2) attention scores ----------------------------
// scores[b,t] = dot(enc[b,t,:], h[b,:]); one wave32 per row, float4 loads.
__global__ void scores_kernel(const float* __restrict__ enc,
                              const float* __restrict__ h,
                              float* __restrict__ scores) {
    int wave = (blockIdx.x * blockDim.x + threadIdx.x) >> 5;
    int lane = threadIdx.x & 31;
    int b = wave / T_;
    int t = wave - b * T_;
    const float4* er = (const float4*)(enc + ((size_t)b * T_ + t) * D_);
    const float4* hr = (const float4*)(h + (size_t)b * D_);
    float acc = 0.f;
    #pragma unroll
    for (int i = 0; i < D_ / 128; ++i) {   // 8 iters * 32 lanes * 4 = 1024
        float4 e = er[i * 32 + lane];
        float4 v = hr[i * 32 + lane];
        acc += e.x * v.x + e.y * v.y + e.z * v.z + e.w * v.w;
    }
    acc = wave_sum(acc);
    if (lane == 0) scores[(size_t)b * T_ + t] = acc;
}

// ------------------ 3) softmax (attn out) + raw-score context --------------
// NOTE: per reference, context uses RAW scores; softmax only feeds attn out.
__global__ void softmax_context_kernel(const float* __restrict__ enc,
                                       const float* __restrict__ scores,
                                       float* __restrict__ attn_out,
                                       float* __restrict__ ctx_out,
                                       float* __restrict__ x) {
    __shared__ float s_raw[T_];
    __shared__ float redmax[8];
    __shared__ float redsum[8];
    int b = blockIdx.x;
    int tid = threadIdx.x;           // 256 threads, T_ == 256
    float s = scores[(size_t)b * T_ + tid];
    s_raw[tid] = s;

    float m = s;
    #pragma unroll
    for (int off = 16; off > 0; off >>= 1) m = fmaxf(m, __shfl_down(m, off, 32));
    if ((tid & 31) == 0) redmax[tid >> 5] = m;
    __syncthreads();
    m = redmax[0];
    #pragma unroll
    for (int i = 1; i < 8; ++i) m = fmaxf(m, redmax[i]);

    float e = expf(s - m);
    float ssum = e;
    #pragma unroll
    for (int off = 16; off > 0; off >>= 1) ssum += __shfl_down(ssum, off, 32);
    if ((tid & 31) == 0) redsum[tid >> 5] = ssum;
    __syncthreads();
    float total = 0.f;
    #pragma unroll
    for (int i = 0; i < 8; ++i) total += redsum[i];

    attn_out[(size_t)b * T_ + tid] = e / total;

    // context[b,d] = sum_t s_raw[t] * enc[b,t,d]; thread owns d = tid + 256*i
    float a0 = 0.f, a1 = 0.f, a2 = 0.f, a3 = 0.f;
    const float* eb = enc + (size_t)b * T_ * D_;
    for (int t = 0; t < T_; ++t) {
        float sc = s_raw[t];
        const float* row = eb + (size_t)t * D_;
        a0 += sc * row[tid];
        a1 += sc * row[tid + 256];
        a2 += sc * row[tid + 512];
        a3 += sc * row[tid + 768];
    }
    float* co = ctx_out + (size_t)b * D_;
    float* xo = x + (size_t)b * DE_;
    co[tid]       = a0;  xo[tid]       = a0;
    co[tid + 256] = a1;  xo[tid + 256] = a1;
    co[tid + 512] = a2;  xo[tid + 512] = a2;
    co[tid + 768] = a3;  xo[tid + 768] = a3;
}

// ------------------------ 4/6) WMMA f32 GEMM -------------------------------
// C(M x N) = A1(M x K1) @ B1 [+ A2(M x K2) @ B2] + bias, row-major f32.
// Block = 8 waves covers a 16 x 512 C tile; each wave owns 16 x 64 (4 v8f
// accumulators). B is streamed through a double-buffered 16 x 512 LDS slab
// via async b128 copies overlapped with WMMA on the previous slab.

// cooperative async copy of one K-slab (lane-guarded for partial N)
__device__ __forceinline__ void copy_slab(const float* __restrict__ Bm, int ldb,
                                          int N, int nblk, int k0, int kcnt,
                                          float* lbuf, int tid) {
    int nchunks = kcnt * (BN / 4);         // float4 chunks, <= 2048
    for (int q = tid; q < nchunks; q += 256) {
        int row  = q >> 7;                 // BN/4 == 128 chunks per row
        int c4   = q & 127;
        int gcol = nblk + c4 * 4;
        if (gcol < N) {                    // N % 4 == 0 -> full chunk in range
            async_copy16(Bm + (size_t)(k0 + row) * ldb + gcol,
                         lbuf + row * BN + c4 * 4);
        }
    }
}

__global__ void __launch_bounds__(256)
wmma_gemm_kernel(const float* __restrict__ A1, int lda1, int K1,
                 const float* __restrict__ A2, int lda2, int K2,
                 const float* __restrict__ B1,
                 const float* __restrict__ B2,
                 int ldb,
                 const float* __restrict__ bias,
                 float* __restrict__ C, int N) {
    __shared__ float sbuf[2][LDSB];        // 2 x 32 KB (of 320 KB/WGP)
    int tid  = threadIdx.x;
    int wave = tid >> 5;
    int lane = tid & 31;
    int m0   = blockIdx.y * 16;
    int nblk = blockIdx.x * BN;
    int n0   = nblk + wave * 64;
    bool active = (n0 < N);                // wave-uniform
    int mlane = lane & 15;
    int khalf = lane >> 4;

    v8f z8 = {0.f, 0.f, 0.f, 0.f, 0.f, 0.f, 0.f, 0.f};
    v8f acc[4] = {z8, z8, z8, z8};

    for (int phase = 0; phase < 2; ++phase) {
        const float* A  = (phase == 0) ? A1 : A2;
        const float* Bm = (phase == 0) ? B1 : B2;
        int lda = (phase == 0) ? lda1 : lda2;
        int K   = (phase == 0) ? K1 : K2;
        if (A == nullptr || K == 0) continue;   // block-uniform
        const float* Arow = A + (size_t)(m0 + mlane) * lda;

        int nslab = (K + BK - 1) / BK;          // K % 4 == 0 always here
        int k1cnt = (K < BK) ? K : BK;
        copy_slab(Bm, ldb, N, nblk, 0, k1cnt, sbuf[0], tid);
        wait_async0();
        __syncthreads();

        int cur = 0;
        for (int s = 0; s < nslab; ++s) {
            int k0   = s * BK;
            int kcnt = K - k0; if (kcnt > BK) kcnt = BK;
            if (s + 1 < nslab) {                // prefetch next slab
                int k0n = k0 + BK;
                int kcn = K - k0n; if (kcn > BK) kcn = BK;
                copy_slab(Bm, ldb, N, nblk, k0n, kcn, sbuf[1 - cur], tid);
            }
            if (active) {                       // wave-uniform: EXEC all-1s
                const float* lb = sbuf[cur] + wave * 64;
                for (int ks = 0; ks < kcnt; ks += 4) {
                    int kk = ks + 2 * khalf;    // A 16x4 f32 fragment layout
                    v2f a;
                    a.x = Arow[k0 + kk];
                    a.y = Arow[k0 + kk + 1];
                    #pragma unroll
                    for (int j = 0; j < 4; ++j) {
                        v2f bf;                 // B 4x16 f32 fragment from LDS
                        bf.x = lb[kk * BN + j * 16 + mlane];
                        bf.y = lb[(kk + 1) * BN + j * 16 + mlane];
                        acc[j] = __builtin_amdgcn_wmma_f32_16x16x4_f32(
                            false, a, false, bf, (short)0, acc[j], false, false);
                    }
                }
            }
            wait_async0();                      // next slab resident
            __syncthreads();                    // all waves done with sbuf[cur]
            cur ^= 1;
        }
    }

    if (!active) return;
    // C layout: VGPR r -> row r + 8*khalf, col = n0 + j*16 + mlane
    #pragma unroll
    for (int j = 0; j < 4; ++j) {
        int col = n0 + j * 16 + mlane;
        float bv = bias ? bias[col] : 0.f;
        #pragma unroll
        for (int r = 0; r < 8; ++r) {
            int row = m0 + r + khalf * 8;
            C[(size_t)row * N + col] = acc[j][r] + bv;
        }
    }
}

// -------------------------- 5) LSTM gate fusion ----------------------------
__global__ void gates_kernel(const float* __restrict__ z,
                             const float* __restrict__ c_old,
                             float* __restrict__ h_new,
                             float* __restrict__ c_new) {
    int i = blockIdx.x * blockDim.x + threadIdx.x;
    if (i >= B_ * D_) return;
    int b = i / D_;
    int d = i - b * D_;
    const float* zr = z + (size_t)b * G4_;
    float zi = zr[d];
    float zf = zr[D_ + d];
    float zg = zr[2 * D_ + d];
    float zo = zr[3 * D_ + d];
    float si = 1.f / (1.f + expf(-zi));
    float sf = 1.f / (1.f + expf(-zf));
    float so = 1.f / (1.f + expf(-zo));
    float c = sf * c_old[i] + si * tanhf(zg);
    c_new[i] = c;
    h_new[i] = so * tanhf(c);
}

// ---------------------------------------------------------------------------
extern "C" void kernel_launch(void* const* d_in, const int* in_sizes, int n_in,
                              void* d_out, int out_size, void* d_ws, size_t ws_size,
                              hipStream_t stream) {
    (void)in_sizes; (void)n_in; (void)out_size; (void)ws_size;

    const int*   idx     = (const int*)  d_in[0];
    const float* enc     = (const float*)d_in[1];
    const float* h_old   = (const float*)d_in[2];
    const float* c_old   = (const float*)d_in[3];
    const float* table   = (const float*)d_in[4];
    const float* W_lstm  = (const float*)d_in[5];
    const float* U_lstm  = (const float*)d_in[6];
    const float* b_lstm  = (const float*)d_in[7];
    const float* W_dense = (const float*)d_in[8];
    const float* b_dense = (const float*)d_in[9];

    // output layout: logits | h_new | c_new | attn | context
    float* out    = (float*)d_out;
    float* logits = out;
    float* h_new  = out + (size_t)B_ * V_;
    float* c_new  = h_new + (size_t)B_ * D_;
    float* attn   = c_new + (size_t)B_ * D_;
    float* ctx    = attn + (size_t)B_ * T_;

    // workspace: scores (B*T) | x (B*(D+E)) | z (B*4D)  ~ 2.9 MB
    float* ws     = (float*)d_ws;
    float* scores = ws;
    float* x      = scores + (size_t)B_ * T_;
    float* z      = x + (size_t)B_ * DE_;

    // 1) embedding gather into x[:, D:]
    emb_gather_kernel<<<(B_ * E_ + 255) / 256, 256, 0, stream>>>(idx, table, x);

    // 2) scores: one wave per (b,t)
    scores_kernel<<<(B_ * T_) / 8, 256, 0, stream>>>(enc, h_old, scores);

    // 3) softmax -> attn; raw-score context -> ctx and x[:, :D]
    softmax_context_kernel<<<B_, 256, 0, stream>>>(enc, scores, attn, ctx, x);

    // 4) z = x @ W_lstm + h @ U_lstm + b_lstm   (M=128, N=4096)
    {
        dim3 grid(G4_ / BN, B_ / 16);   // (8, 8)
        wmma_gemm_kernel<<<grid, 256, 0, stream>>>(
            x, DE_, DE_, h_old, D_, D_, W_lstm, U_lstm, G4_, b_lstm, z, G4_);
    }

    // 5) gates -> h_new, c_new
    gates_kernel<<<(B_ * D_ + 255) / 256, 256, 0, stream>>>(z, c_old, h_new, c_new);

    // 6) logits = h_new @ W_dense + b_dense   (M=128, K=1024, N=32000)
    {
        dim3 grid((V_ + BN - 1) / BN, B_ / 16);  // (63, 8)
        wmma_gemm_kernel<<<grid, 256, 0, stream>>>(
            h_new, D_, D_, nullptr, 0, 0, W_dense, nullptr, V_, b_dense, logits, V_);
    }
}